// Rea_sentence_80135499809389
// MI455X (gfx1250) — compile-verified
//
#include <hip/hip_runtime.h>

typedef __attribute__((ext_vector_type(2))) float v2f;
typedef __attribute__((ext_vector_type(8))) float v8f;

#define BB 4
#define HH 16
#define NN 1024
#define DDIM 768
#define HD 48

// Kernel 1: fused threshold + section-mask + softmax + (att @ xv).
// One wave32 per (b, h, 16-row q tile). Streams the 268MB score tensor once.
__global__ __launch_bounds__(32)
void attn_ctx_kernel(const float* __restrict__ x,      // [B,N,D]
                     const float* __restrict__ smask,  // [N,N]
                     const float* __restrict__ score,  // [B,H,N,N]
                     const float* __restrict__ thrp,   // [1]
                     float* __restrict__ ctx)          // [B,N,D] (heads merged)
{
    const int lane  = threadIdx.x;           // 0..31
    const int wid   = blockIdx.x;            // 0 .. B*H*64-1
    const int qt    = wid & 63;
    const int h     = (wid >> 6) & (HH - 1);
    const int b     = wid >> 10;
    const int q0    = qt << 4;
    const int mrow  = lane & 15;             // A-matrix row (M) for this lane
    const int khalf = (lane >> 4) << 1;      // K sub-offset: 0 (lanes 0-15) or 2 (16-31)
    const float thr = thrp[0];

    // A-layout source rows (per-lane row pointers)
    const float* srow  = score + ((((size_t)b * HH + h) * NN) + q0 + mrow) * NN;
    const float* gmask = smask + (size_t)(q0 + mrow) * NN;
    // B-layout source column base: xv[k][d] = x[b, k, h*48 + d]; col = h*48 + 16t + (lane&15)
    const float* xcol  = x + (size_t)b * NN * DDIM + h * HD + (lane & 15);

    v8f acc0 = {}; v8f acc1 = {}; v8f acc2 = {};
    float rs = 0.0f;   // partial row-sum of exp() for this lane's (row, K-half)

    for (int kc = 0; kc < NN; kc += 4) {
        const int k0 = kc + khalf;
        float s0 = srow[k0];
        float s1 = srow[k0 + 1];
        const float m0 = gmask[k0];
        const float m1 = gmask[k0 + 1];
        s0 = (s0 < thr) ? 0.0f : s0;
        s1 = (s1 < thr) ? 0.0f : s1;
        const float l0 = m0 + s0;
        const float l1 = m1 + s1;
        // masked_fill(-1e9) + softmax  ==  p = (l==0) ? 0 : exp(l)   (no max needed: l <= ~8)
        const float p0 = (l0 == 0.0f) ? 0.0f : __expf(l0);
        const float p1 = (l1 == 0.0f) ? 0.0f : __expf(l1);
        rs += p0 + p1;

        v2f a; a[0] = p0; a[1] = p1;       // A 16x4 f32 layout

        const float* xr = xcol + (size_t)k0 * DDIM;   // row k0 of xv, this lane's column
        v2f b0, b1, b2;                    // B 4x16 layout for the 3 n-tiles (HD=48)
        b0[0] = xr[0];   b0[1] = xr[DDIM];
        b1[0] = xr[16];  b1[1] = xr[DDIM + 16];
        b2[0] = xr[32];  b2[1] = xr[DDIM + 32];

        acc0 = __builtin_amdgcn_wmma_f32_16x16x4_f32(false, a, false, b0, (short)0, acc0, false, false);
        acc1 = __builtin_amdgcn_wmma_f32_16x16x4_f32(false, a, false, b1, (short)0, acc1, false, false);
        acc2 = __builtin_amdgcn_wmma_f32_16x16x4_f32(false, a, false, b2, (short)0, acc2, false, false);
    }

    // combine the two K-halves: after this, lane L holds rowsum for row (L & 15)
    rs += __shfl_xor(rs, 16, 32);
    const float invrs = 1.0f / rs;

    // D layout: lane L holds column n = L&15, rows m = v + 8*(L>>4)
    float* cb = ctx + ((size_t)b * NN + q0) * DDIM + h * HD + (lane & 15);
#pragma unroll
    for (int v = 0; v < 8; ++v) {
        const int m = v + ((lane >> 4) << 3);
        const float inv = __shfl(invrs, m, 32);   // rowsum^-1 for row m lives in lane m
        cb[(size_t)m * DDIM + 0]  = acc0[v] * inv;
        cb[(size_t)m * DDIM + 16] = acc1[v] * inv;
        cb[(size_t)m * DDIM + 32] = acc2[v] * inv;
    }
}

// Kernel 2: out[i, n] = sum_k ctx[i, k] * W3[n, k] + b3[n]   (torch Linear: ctx @ W3^T + b3)
// One wave32 per 16x16 output tile, K = 768.
__global__ __launch_bounds__(32)
void proj_kernel(const float* __restrict__ ctx,  // [B*N, D] = [4096, 768]
                 const float* __restrict__ W3,   // [D, D]
                 const float* __restrict__ b3,   // [D]
                 float* __restrict__ out)        // [4096, 768]
{
    const int lane  = threadIdx.x;
    const int wid   = blockIdx.x;                // 0 .. 256*48-1
    const int nt    = wid % (DDIM / 16);
    const int it    = wid / (DDIM / 16);
    const int i0    = it << 4;
    const int n0    = nt << 4;
    const int r     = lane & 15;
    const int khalf = (lane >> 4) << 1;

    // A: lane L -> ctx row (i0 + L&15), K = kc + khalf + v  (contiguous pair -> v2f load)
    const float* arow = ctx + (size_t)(i0 + r) * DDIM + khalf;
    // B: B[k][n] = W3[n][k] -> lane L reads W3 row (n0 + L&15), contiguous K pair
    const float* brow = W3  + (size_t)(n0 + r) * DDIM + khalf;

    v8f acc = {};
#pragma unroll 4
    for (int kc = 0; kc < DDIM; kc += 4) {
        v2f a    = *(const v2f*)(arow + kc);
        v2f bmat = *(const v2f*)(brow + kc);
        acc = __builtin_amdgcn_wmma_f32_16x16x4_f32(false, a, false, bmat, (short)0, acc, false, false);
    }

    const float bias = b3[n0 + r];
    float* ob = out + (size_t)i0 * DDIM + n0 + r;
#pragma unroll
    for (int v = 0; v < 8; ++v) {
        const int m = v + ((lane >> 4) << 3);
        ob[(size_t)m * DDIM] = acc[v] + bias;
    }
}

extern "C" void kernel_launch(void* const* d_in, const int* in_sizes, int n_in,
                              void* d_out, int out_size, void* d_ws, size_t ws_size,
                              hipStream_t stream) {
    const float* x     = (const float*)d_in[0];   // [B,N,D]
    const float* smask = (const float*)d_in[1];   // [N,N]
    const float* score = (const float*)d_in[2];   // [B,H,N,N]
    const float* thr   = (const float*)d_in[3];   // [1]
    const float* W3    = (const float*)d_in[4];   // [D,D]
    const float* b3    = (const float*)d_in[5];   // [D]
    float* out = (float*)d_out;                   // [B,N,D]
    float* ctx = (float*)d_ws;                    // needs B*N*D*4 = 12.6 MB scratch

    attn_ctx_kernel<<<BB * HH * (NN / 16), 32, 0, stream>>>(x, smask, score, thr, ctx);
    proj_kernel<<<(BB * NN / 16) * (DDIM / 16), 32, 0, stream>>>(ctx, W3, b3, out);
}